// SAMixtureConv2d_16338055594621
// MI455X (gfx1250) — compile-verified
//
#include <hip/hip_runtime.h>
#include <math.h>

// ---- problem constants (from reference) ----
#define BATCH   2
#define CIN     3
#define HH      224
#define WWID    224
#define COUT    64
#define KH      7
#define KW      7
#define STR     2
#define PADP    3
#define PH      230
#define PW      230
#define FH      112
#define FW      112
#define NPIX    (PH*PW)        // 52900 padded pixels
#define NQ      (FH*FW)        // 12544 queries
#define NK      (KH*KW)        // 49 window taps
#define MIX     4

typedef __attribute__((ext_vector_type(2))) float v2f;
typedef __attribute__((ext_vector_type(8))) float v8f;

// =====================================================================
// K0: per-padded-pixel mixture weights
//   p_abm[m][i][j] = softmax_m( sum_c (ex[c,j]+ey[c,i]) * em[m,c] )
// =====================================================================
__global__ __launch_bounds__(256) void k_pabm(const float* __restrict__ ex,
                                              const float* __restrict__ ey,
                                              const float* __restrict__ em,
                                              float* __restrict__ pabm) {
  unsigned t = blockIdx.x * 256u + threadIdx.x;
  if (t >= NPIX) return;
  unsigned i = t / PW, j = t % PW;
  float s[MIX] = {0.f, 0.f, 0.f, 0.f};
  #pragma unroll 4
  for (unsigned c = 0; c < COUT; ++c) {
    float e = ex[c * PW + j] + ey[c * PH + i];
    #pragma unroll
    for (int m = 0; m < MIX; ++m) s[m] += e * em[m * COUT + c];
  }
  float mx = fmaxf(fmaxf(s[0], s[1]), fmaxf(s[2], s[3]));
  float sum = 0.f;
  #pragma unroll
  for (int m = 0; m < MIX; ++m) { s[m] = __expf(s[m] - mx); sum += s[m]; }
  float inv = 1.f / sum;
  #pragma unroll
  for (int m = 0; m < MIX; ++m) pabm[m * NPIX + t] = s[m] * inv;
}

// =====================================================================
// K1: fused q/k/v projection via V_WMMA_F32_16X16X4_F32.
// One wave = one tile of 16 consecutive padded pixels (one row) x 64 channels.
// A = 16 pixels x 4 (CIN=3 padded to K=4), B = 4 x 16 weight chunk.
// q/k and 4 value mixtures -> 24 WMMAs per tile; value mixtures combined
// with per-pixel p_abm weights in VALU. Outputs stored channel-innermost.
// =====================================================================
#define JT      15                    // ceil(230/16) j-tiles per row
#define NWAVES  (BATCH * PH * JT)     // 6900 waves

__global__ __launch_bounds__(256) void k_project(const float* __restrict__ x,
                                                 const float* __restrict__ wq,
                                                 const float* __restrict__ wk,
                                                 const float* __restrict__ wv,
                                                 const float* __restrict__ pabm,
                                                 float* __restrict__ qc,   // [b][fi][fj][64]
                                                 float* __restrict__ vkt,  // [b][i][j][64]
                                                 float* __restrict__ vvt)  // [b][i][j][64]
{
  int wave = blockIdx.x * 8 + (threadIdx.x >> 5);
  if (wave >= NWAVES) return;                    // wave-uniform exit, EXEC stays full
  int lane = threadIdx.x & 31;
  int half = lane >> 4;                          // 0: K=0/1, 1: K=2/3
  int ln   = lane & 15;

  int b   = wave / (PH * JT);
  int rem = wave % (PH * JT);
  int i   = rem / JT;
  int j0  = (rem % JT) * 16;

  // ---- A fragment: lane row M = ln (pixel j0+ln), K = 2*half, 2*half+1 ----
  int ja = j0 + ln;
  int ui = i - PADP, uj = ja - PADP;
  bool inb = (ja < PW) && (ui >= 0) && (ui < HH) && (uj >= 0) && (uj < WWID);
  const float* xb = x + (unsigned)b * (CIN * HH * WWID);
  int c0 = 2 * half;                             // 0 or 2 (both < CIN)
  unsigned xoff = (unsigned)(ui * WWID + uj);
  v2f A;
  A.x = inb ? xb[(unsigned)c0 * (HH * WWID) + xoff] : 0.f;
  A.y = (inb && half == 0) ? xb[(unsigned)(HH * WWID) + xoff] : 0.f; // K=3 pad -> 0

  // ---- per-element mixture weights (element r -> pixel p = r + 8*half) ----
  float pm[8][MIX];
  #pragma unroll
  for (int r = 0; r < 8; ++r) {
    int p  = r + 8 * half;
    int jj = j0 + p;
    bool ok = jj < PW;
    unsigned idx = (unsigned)(i * PW + jj);
    #pragma unroll
    for (int m = 0; m < MIX; ++m) pm[r][m] = ok ? pabm[(unsigned)m * NPIX + idx] : 0.f;
  }

  #pragma unroll
  for (int nb = 0; nb < 4; ++nb) {
    int N = nb * 16 + ln;                        // output channel of this lane
    // ---- B fragments: rows K = 2*half (valid), 2*half+1 (1 valid, 3 -> 0) ----
    v2f Bq, Bk, Bv[MIX];
    Bq.x = wq[(unsigned)(c0 * COUT + N)];  Bq.y = half ? 0.f : wq[(unsigned)(COUT + N)];
    Bk.x = wk[(unsigned)(c0 * COUT + N)];  Bk.y = half ? 0.f : wk[(unsigned)(COUT + N)];
    #pragma unroll
    for (int m = 0; m < MIX; ++m) {
      Bv[m].x = wv[(unsigned)((m * CIN + c0) * COUT + N)];
      Bv[m].y = half ? 0.f : wv[(unsigned)((m * CIN + 1) * COUT + N)];
    }

    v8f z = {};
    v8f Cq  = __builtin_amdgcn_wmma_f32_16x16x4_f32(false, A, false, Bq,    (short)0, z, false, false);
    v8f Ck  = __builtin_amdgcn_wmma_f32_16x16x4_f32(false, A, false, Bk,    (short)0, z, false, false);
    v8f Cv0 = __builtin_amdgcn_wmma_f32_16x16x4_f32(false, A, false, Bv[0], (short)0, z, false, false);
    v8f Cv1 = __builtin_amdgcn_wmma_f32_16x16x4_f32(false, A, false, Bv[1], (short)0, z, false, false);
    v8f Cv2 = __builtin_amdgcn_wmma_f32_16x16x4_f32(false, A, false, Bv[2], (short)0, z, false, false);
    v8f Cv3 = __builtin_amdgcn_wmma_f32_16x16x4_f32(false, A, false, Bv[3], (short)0, z, false, false);

    // ---- scatter results: element r belongs to pixel (i, j0 + r + 8*half) ----
    #pragma unroll
    for (int r = 0; r < 8; ++r) {
      int p  = r + 8 * half;
      int jj = j0 + p;
      if (jj < PW) {
        unsigned base = (unsigned)(((b * PH + i) * PW + jj) * COUT + N);
        vkt[base] = Ck[r];
        vvt[base] = pm[r][0] * Cv0[r] + pm[r][1] * Cv1[r]
                  + pm[r][2] * Cv2[r] + pm[r][3] * Cv3[r];
        // query centers: (i,j) = (3+2fi, 3+2fj)
        if ((i & 1) && i >= PADP && i <= PADP + STR * (FH - 1) &&
            (jj & 1) && jj >= PADP && jj <= PADP + STR * (FW - 1)) {
          int fi = (i - PADP) >> 1, fj = (jj - PADP) >> 1;
          qc[(unsigned)(((b * FH + fi) * FW + fj) * COUT + N)] = Cq[r];
        }
      }
    }
  }
}

// =====================================================================
// K2: attention scores.  score[b,k,n] = sum_o q[o,n] * (win_k[o,k,n] + rel[o,k])
// rel staged in LDS ([k][64], float4-readable). One thread per query.
// =====================================================================
__global__ __launch_bounds__(256) void k_scores(const float* __restrict__ qc,
                                                const float* __restrict__ vkt,
                                                const float* __restrict__ rx,
                                                const float* __restrict__ ry,
                                                float* __restrict__ scores) {
  __shared__ float rel[NK * COUT];
  for (unsigned idx = threadIdx.x; idx < NK * COUT; idx += 256) {
    unsigned o = idx & 63, k = idx >> 6;
    unsigned kh = k / KW, kw = k % KW;
    rel[idx] = (o < 32) ? rx[o * KW + kw] : ry[(o - 32) * KH + kh];
  }
  __syncthreads();

  unsigned t = blockIdx.x * 256u + threadIdx.x;  // grid sized exactly BATCH*NQ
  unsigned b = t / NQ, n = t % NQ;
  unsigned fi = n / FW, fj = n % FW;

  float4 q[16];
  const float4* qp = (const float4*)(qc + (unsigned)(((b * FH + fi) * FW + fj) * COUT));
  #pragma unroll
  for (int o4 = 0; o4 < 16; ++o4) q[o4] = qp[o4];

  unsigned rowbase = (b * PH + STR * fi) * PW + STR * fj;   // element row of window origin
  for (int kh = 0; kh < KH; ++kh) {
    // prefetch next window row into cache hierarchy (L2-resident working set)
    if (kh + 1 < KH)
      __builtin_prefetch(vkt + (rowbase + (unsigned)(kh + 1) * PW) * COUT, 0, 3);
    for (int kw = 0; kw < KW; ++kw) {
      unsigned k = (unsigned)(kh * KW + kw);
      const float4* kp = (const float4*)(vkt + (rowbase + (unsigned)kh * PW + (unsigned)kw) * COUT);
      const float4* rp = (const float4*)(&rel[k * COUT]);
      float acc = 0.f;
      #pragma unroll
      for (int o4 = 0; o4 < 16; ++o4) {
        float4 kv = kp[o4], rl = rp[o4], qq = q[o4];
        acc += qq.x * (kv.x + rl.x) + qq.y * (kv.y + rl.y)
             + qq.z * (kv.z + rl.z) + qq.w * (kv.w + rl.w);
      }
      scores[(b * NK + k) * NQ + n] = acc;
    }
  }
}

// =====================================================================
// K3: softmax row stats over the spatial axis (12544 per (b,k) row).
// One workgroup per row; LDS tree reduction for max and sum(exp).
// =====================================================================
__global__ __launch_bounds__(256) void k_rowstats(const float* __restrict__ scores,
                                                  float* __restrict__ rmax,
                                                  float* __restrict__ rsum) {
  __shared__ float red[256];
  const float* s = scores + (unsigned)blockIdx.x * NQ;
  float mx = -INFINITY;
  for (unsigned n = threadIdx.x; n < NQ; n += 256) mx = fmaxf(mx, s[n]);
  red[threadIdx.x] = mx; __syncthreads();
  for (int st = 128; st > 0; st >>= 1) {
    if (threadIdx.x < (unsigned)st) red[threadIdx.x] = fmaxf(red[threadIdx.x], red[threadIdx.x + st]);
    __syncthreads();
  }
  mx = red[0]; __syncthreads();
  float sm = 0.f;
  for (unsigned n = threadIdx.x; n < NQ; n += 256) sm += __expf(s[n] - mx);
  red[threadIdx.x] = sm; __syncthreads();
  for (int st = 128; st > 0; st >>= 1) {
    if (threadIdx.x < (unsigned)st) red[threadIdx.x] += red[threadIdx.x + st];
    __syncthreads();
  }
  if (threadIdx.x == 0) { rmax[blockIdx.x] = mx; rsum[blockIdx.x] = red[0]; }
}

// =====================================================================
// K4: output = sum_k attn[b,k,n] * win_v[b,:,k,n] + bias.  One thread per query.
// =====================================================================
__global__ __launch_bounds__(256) void k_output(const float* __restrict__ scores,
                                                const float* __restrict__ rmax,
                                                const float* __restrict__ rsum,
                                                const float* __restrict__ vvt,
                                                const float* __restrict__ bias,
                                                float* __restrict__ out) {
  __shared__ float smx[BATCH * NK], sinv[BATCH * NK];
  for (unsigned idx = threadIdx.x; idx < BATCH * NK; idx += 256) {
    smx[idx] = rmax[idx];
    sinv[idx] = 1.f / rsum[idx];
  }
  __syncthreads();

  unsigned t = blockIdx.x * 256u + threadIdx.x;  // grid sized exactly BATCH*NQ
  unsigned b = t / NQ, n = t % NQ;
  unsigned fi = n / FW, fj = n % FW;

  float4 acc[16];
  #pragma unroll
  for (int o4 = 0; o4 < 16; ++o4) acc[o4] = make_float4(0.f, 0.f, 0.f, 0.f);

  unsigned rowbase = (b * PH + STR * fi) * PW + STR * fj;
  for (int kh = 0; kh < KH; ++kh) {
    if (kh + 1 < KH)
      __builtin_prefetch(vvt + (rowbase + (unsigned)(kh + 1) * PW) * COUT, 0, 3);
    for (int kw = 0; kw < KW; ++kw) {
      unsigned k = (unsigned)(kh * KW + kw);
      float s = scores[(b * NK + k) * NQ + n];
      float w = __expf(s - smx[b * NK + k]) * sinv[b * NK + k];
      const float4* vp = (const float4*)(vvt + (rowbase + (unsigned)kh * PW + (unsigned)kw) * COUT);
      #pragma unroll
      for (int o4 = 0; o4 < 16; ++o4) {
        float4 v = vp[o4];
        acc[o4].x += w * v.x; acc[o4].y += w * v.y;
        acc[o4].z += w * v.z; acc[o4].w += w * v.w;
      }
    }
  }
  const float* accf = (const float*)acc;
  #pragma unroll 8
  for (int o = 0; o < COUT; ++o) {
    out[(unsigned)(((b * COUT + o) * FH + fi) * FW + fj)] = accf[o] + bias[o];
  }
}

// =====================================================================
extern "C" void kernel_launch(void* const* d_in, const int* in_sizes, int n_in,
                              void* d_out, int out_size, void* d_ws, size_t ws_size,
                              hipStream_t stream) {
  const float* x    = (const float*)d_in[0];
  const float* wq   = (const float*)d_in[1];
  const float* wk   = (const float*)d_in[2];
  const float* wv   = (const float*)d_in[3];
  const float* rx   = (const float*)d_in[4];
  const float* ry   = (const float*)d_in[5];
  const float* ex   = (const float*)d_in[6];
  const float* ey   = (const float*)d_in[7];
  const float* em   = (const float*)d_in[8];
  const float* bias = (const float*)d_in[9];
  float* out = (float*)d_out;

  char* ws = (char*)d_ws;
  float* pabm   = (float*)ws; ws += sizeof(float) * MIX * NPIX;                  // 846 KB
  float* qc     = (float*)ws; ws += sizeof(float) * BATCH * NQ * COUT;           // 6.4 MB
  float* vkt    = (float*)ws; ws += sizeof(float) * (size_t)BATCH * NPIX * COUT; // 27 MB
  float* vvt    = (float*)ws; ws += sizeof(float) * (size_t)BATCH * NPIX * COUT; // 27 MB
  float* scores = (float*)ws; ws += sizeof(float) * (size_t)BATCH * NK * NQ;     // 4.9 MB
  float* rmax   = (float*)ws; ws += sizeof(float) * BATCH * NK;
  float* rsum   = (float*)ws; ws += sizeof(float) * BATCH * NK;

  k_pabm<<<(NPIX + 255) / 256, 256, 0, stream>>>(ex, ey, em, pabm);
  k_project<<<(NWAVES + 7) / 8, 256, 0, stream>>>(x, wq, wk, wv, pabm, qc, vkt, vvt);
  k_scores<<<(BATCH * NQ) / 256, 256, 0, stream>>>(qc, vkt, rx, ry, scores);
  k_rowstats<<<BATCH * NK, 256, 0, stream>>>(scores, rmax, rsum);
  k_output<<<(BATCH * NQ) / 256, 256, 0, stream>>>(scores, rmax, rsum, vvt, bias, out);
}